// PairInteractionBlock_78365973283131
// MI455X (gfx1250) — compile-verified
//
#include <hip/hip_runtime.h>
#include <math.h>

// ---------------------------------------------------------------------------
// PairInteractionBlock on MI455X (gfx1250), fp32 WMMA 16x16x4 everywhere.
// HBM-bound (~600MB moved, ~54 GFLOP) -> keep full f32 precision; fuse
// radial-MLP + tensor-product + channel-mixing so the (E,576) weight tensor
// never touches HBM. Weights are pre-packed into the wave32 WMMA B-register
// layout (row-pair interleaved, fan-in scale folded in) so every B operand is
// a single aligned global_load_b64.
// ---------------------------------------------------------------------------

typedef float v2f __attribute__((ext_vector_type(2)));
typedef float v8f __attribute__((ext_vector_type(8)));

__device__ __forceinline__ v8f wmma4(v2f a, v2f b, v8f c) {
  // D(16x16,f32) = A(16x4,f32) * B(4x16,f32) + C
  return __builtin_amdgcn_wmma_f32_16x16x4_f32(
      /*neg_a=*/false, a, /*neg_b=*/false, b,
      /*c_mod=*/(short)0, c, /*reuse_a=*/false, /*reuse_b=*/false);
}

// Packed-B fetch: P holds W[K][N] as row pairs: P[((2k+h1)*N + n)*2 + {0,1}]
// = W[4k + 2*h1 + {0,1}][n].  One 8B load per WMMA B operand.
__device__ __forceinline__ v2f ldBp(const float* __restrict__ P, int N, int k,
                                    int h1, int ncol) {
  return *(const v2f*)(P + ((((k << 1) + h1) * N + ncol) << 1));
}

__device__ __forceinline__ float ssp(float x) {
  // softplus(x) - log(2), numerically stable
  float ax = fabsf(x);
  return fmaxf(x, 0.0f) + __logf(1.0f + __expf(-ax)) - 0.6931471805599453f;
}

#define DIMF 160

// ============================================================================
// Weight pre-pack: dst[((r>>1)*N + n)*2 + (r&1)] = src[r*N + n] * scale
// ============================================================================
__global__ void __launch_bounds__(256)
pack_b_kernel(const float* __restrict__ src, float* __restrict__ dst,
              int K, int N, float scale) {
  int idx = blockIdx.x * blockDim.x + threadIdx.x;
  if (idx >= K * N) return;
  int r = idx / N, n = idx - r * N;
  dst[(((r >> 1) * N + n) << 1) + (r & 1)] = src[idx] * scale;
}

// ============================================================================
// Kernel 1: up_src / up_tar = o3_linear(node_feats, W{src,tar}{0,1})
// one wave handles a 16-node tile, both transforms. Scales folded into B.
// ============================================================================
__global__ void __launch_bounds__(128)
node_transform_kernel(const float* __restrict__ node_feats,
                      const float* __restrict__ pWsrc0, const float* __restrict__ pWsrc1,
                      const float* __restrict__ pWtar0, const float* __restrict__ pWtar1,
                      float* __restrict__ up_src, float* __restrict__ up_tar,
                      int n_nodes) {
  const int lane = threadIdx.x & 31;
  const int wave = threadIdx.x >> 5;
  const int tile = blockIdx.x * (blockDim.x >> 5) + wave;
  const int node0 = tile * 16;
  if (node0 >= n_nodes) return;

  const int l15  = lane & 15;
  const int h1   = lane >> 4;
  const int h2   = h1 << 1;  // A/B K sub-offset for this half-wave
  const int mrow = h1 << 3;  // D row offset for this half-wave
  const bool full = (node0 + 16 <= n_nodes);

  const int arow = min(node0 + l15, n_nodes - 1);
  const float* nf = node_feats + (long)arow * DIMF;

  v2f aS[16];
#pragma unroll
  for (int k = 0; k < 16; ++k) aS[k] = *(const v2f*)(nf + 4 * k + h2);
  v2f aV[3][8];
#pragma unroll
  for (int i = 0; i < 3; ++i)
#pragma unroll
    for (int k = 0; k < 8; ++k) {
      int c = 4 * k + h2;
      aV[i][k].x = nf[64 + 3 * c + i];
      aV[i][k].y = nf[64 + 3 * (c + 1) + i];
    }

#pragma unroll 1
  for (int t = 0; t < 2; ++t) {
    const float* W0 = t ? pWtar0 : pWsrc0;
    const float* W1 = t ? pWtar1 : pWsrc1;
    float* up = t ? up_tar : up_src;

    // scalar: (16x64) @ (64x64)
#pragma unroll
    for (int n = 0; n < 4; ++n) {
      v8f acc = {};
#pragma unroll
      for (int k = 0; k < 16; ++k)
        acc = wmma4(aS[k], ldBp(W0, 64, k, h1, n * 16 + l15), acc);
      float* o = up + (long)(node0 + mrow) * DIMF + n * 16 + l15;
      if (full) {
#pragma unroll
        for (int j = 0; j < 8; ++j) o[(long)j * DIMF] = acc[j];
      } else {
#pragma unroll
        for (int j = 0; j < 8; ++j)
          if (node0 + j + mrow < n_nodes) o[(long)j * DIMF] = acc[j];
      }
    }
    // vector: per component, (16x32) @ (32x32)
#pragma unroll
    for (int i = 0; i < 3; ++i)
#pragma unroll
      for (int n = 0; n < 2; ++n) {
        v8f acc = {};
#pragma unroll
        for (int k = 0; k < 8; ++k)
          acc = wmma4(aV[i][k], ldBp(W1, 32, k, h1, n * 16 + l15), acc);
        float* o = up + (long)(node0 + mrow) * DIMF + 64 + 3 * (n * 16 + l15) + i;
        if (full) {
#pragma unroll
          for (int j = 0; j < 8; ++j) o[(long)j * DIMF] = acc[j];
        } else {
#pragma unroll
          for (int j = 0; j < 8; ++j)
            if (node0 + j + mrow < n_nodes) o[(long)j * DIMF] = acc[j];
        }
      }
  }
}

// ============================================================================
// Kernel 2: fused per-edge pipeline, one wave per 16-edge tile.
//   MLP (WMMA) -> w_b tiles (WMMA, per branch) -> TP elementwise into S/V (LDS)
//   -> mixing + skip GEMMs (WMMA) -> out
// All LDS buffers are wave-private (blockDim == 32): DS ops are in-order
// within a wave, so no barriers are required. Strides kept even so A-operand
// LDS reads are ds_load_b64.
// ============================================================================
#define H_LD 68
#define W_LD 196
#define S_LD 100
#define LDS_FLOATS (16 * H_LD + 16 * W_LD + 16 * S_LD + 3 * 16 * S_LD)

__global__ void __launch_bounds__(32)
edge_fused_kernel(const int* __restrict__ edge_index,
                  const float* __restrict__ edge_feats,
                  const float* __restrict__ edge_attrs,
                  const float* __restrict__ edge_embed,
                  const float* __restrict__ pWskip0, const float* __restrict__ pWskip1,
                  const float* __restrict__ pWm1, const float* __restrict__ pWm2,
                  const float* __restrict__ pWm3, const float* __restrict__ pWm4,
                  const float* __restrict__ pWc0, const float* __restrict__ pWc1,
                  const float* __restrict__ up_src, const float* __restrict__ up_tar,
                  float* __restrict__ out, int n_edges) {
  __shared__ float lds[LDS_FLOATS];

  const int lane = threadIdx.x & 31;
  const int e0 = blockIdx.x * 16;
  if (e0 >= n_edges) return;

  float* hT  = lds;                 // 16 x H_LD
  float* wbT = hT + 16 * H_LD;      // 16 x W_LD (one branch's 192 weights)
  float* ST  = wbT + 16 * W_LD;     // 16 x S_LD
  float* VT  = ST + 16 * S_LD;      // 3 planes of 16 x S_LD

  const int l15  = lane & 15;
  const int h1   = lane >> 4;
  const int h2   = h1 << 1;
  const int mrow = h1 << 3;
  const bool full = (e0 + 16 <= n_edges);

  const int earow = min(e0 + l15, n_edges - 1);

  // ---------------- Phase A: radial MLP ----------------
  {
    // h1 = ssp(embed(16x8) @ Wm1(8x64))
    const float* eb = edge_embed + (long)earow * 8;
    v2f aE[2];
#pragma unroll
    for (int k = 0; k < 2; ++k) aE[k] = *(const v2f*)(eb + 4 * k + h2);
#pragma unroll
    for (int n = 0; n < 4; ++n) {
      v8f acc = {};
#pragma unroll
      for (int k = 0; k < 2; ++k)
        acc = wmma4(aE[k], ldBp(pWm1, 64, k, h1, n * 16 + l15), acc);
#pragma unroll
      for (int j = 0; j < 8; ++j)
        hT[(j + mrow) * H_LD + n * 16 + l15] = ssp(acc[j]);
    }
  }
  // h = ssp(h @ Wm2), h = ssp(h @ Wm3)
#pragma unroll 1
  for (int layer = 0; layer < 2; ++layer) {
    const float* Wm = layer ? pWm3 : pWm2;
    v2f aH[16];
#pragma unroll
    for (int k = 0; k < 16; ++k)
      aH[k] = *(const v2f*)(hT + l15 * H_LD + 4 * k + h2);
    v8f acc[4] = {};
#pragma unroll
    for (int n = 0; n < 4; ++n)
#pragma unroll
      for (int k = 0; k < 16; ++k)
        acc[n] = wmma4(aH[k], ldBp(Wm, 64, k, h1, n * 16 + l15), acc[n]);
#pragma unroll
    for (int n = 0; n < 4; ++n)
#pragma unroll
      for (int j = 0; j < 8; ++j)
        hT[(j + mrow) * H_LD + n * 16 + l15] = ssp(acc[n][j]);
  }

  // cache h3 in A layout for the three Wm4 branch GEMMs
  v2f aH3[16];
#pragma unroll
  for (int k = 0; k < 16; ++k)
    aH3[k] = *(const v2f*)(hT + l15 * H_LD + 4 * k + h2);

  // zero S and V accumulators (covers padded rows too)
  for (int idx = lane; idx < 16 * S_LD; idx += 32) {
    ST[idx] = 0.0f;
    VT[idx] = 0.0f;
    VT[16 * S_LD + idx] = 0.0f;
    VT[32 * S_LD + idx] = 0.0f;
  }

  // ---------------- Phase B: per-branch w tile + tensor product ----------------
  const float inv_sqrt3 = 0.5773502691896258f;
#pragma unroll 1
  for (int b = 0; b < 3; ++b) {
    // w_b(16x192) = h3(16x64) @ Wm4[:, b*192:(b+1)*192]
#pragma unroll
    for (int n = 0; n < 12; ++n) {
      v8f acc = {};
#pragma unroll
      for (int k = 0; k < 16; ++k)
        acc = wmma4(aH3[k], ldBp(pWm4, 576, k, h1, b * 192 + n * 16 + l15), acc);
#pragma unroll
      for (int j = 0; j < 8; ++j)
        wbT[(j + mrow) * W_LD + n * 16 + l15] = acc[j];
    }

    // tensor product: accumulate S (16x96) and V (3 x 16x96)
    for (int idx = lane; idx < 16 * 96; idx += 32) {
      int el = idx / 96, c = idx - el * 96;
      int e = e0 + el;
      bool valid = e < n_edges;
      int ec = valid ? e : (n_edges - 1);
      const float* F;
      if (b == 0)
        F = up_src + (long)edge_index[ec] * DIMF;
      else if (b == 1)
        F = up_tar + (long)edge_index[(long)n_edges + ec] * DIMF;
      else
        F = edge_feats + (long)ec * DIMF;
      const float* ya = edge_attrs + (long)ec * 4;
      float y0 = ya[0];
      if (c < 64) {
        float s   = F[c];
        float w00 = wbT[el * W_LD + c];
        float w01 = wbT[el * W_LD + 96 + c];
        if (valid) {
          ST[el * S_LD + c] += w00 * s * y0;
          float ws = w01 * s;
          VT[el * S_LD + c] += ws * ya[1];
          VT[16 * S_LD + el * S_LD + c] += ws * ya[2];
          VT[32 * S_LD + el * S_LD + c] += ws * ya[3];
        }
      } else {
        int kk = c - 64;
        float v0 = F[64 + 3 * kk + 0];
        float v1 = F[64 + 3 * kk + 1];
        float v2 = F[64 + 3 * kk + 2];
        float wdot = wbT[el * W_LD + 64 + kk];
        float w10  = wbT[el * W_LD + 160 + kk];
        if (valid) {
          ST[el * S_LD + c] += wdot * (v0 * ya[1] + v1 * ya[2] + v2 * ya[3]) * inv_sqrt3;
          VT[el * S_LD + c] += w10 * v0 * y0;
          VT[16 * S_LD + el * S_LD + c] += w10 * v1 * y0;
          VT[32 * S_LD + el * S_LD + c] += w10 * v2 * y0;
        }
      }
    }
  }

  // ---------------- Phase C: channel mixing + skip connection ----------------
  // (1/sqrt(96), 1/8, 1/sqrt(32) are folded into pWc0/pWc1/pWskip0/pWskip1)
  const float* ef = edge_feats + (long)earow * DIMF;

  // scalar out[:, 0:64] = S@Wc0 + s_feat@Wskip0
  {
    v8f acc[4] = {};
#pragma unroll
    for (int k = 0; k < 24; ++k) {
      v2f a = *(const v2f*)(ST + l15 * S_LD + 4 * k + h2);
#pragma unroll
      for (int n = 0; n < 4; ++n)
        acc[n] = wmma4(a, ldBp(pWc0, 64, k, h1, n * 16 + l15), acc[n]);
    }
#pragma unroll
    for (int k = 0; k < 16; ++k) {
      v2f a = *(const v2f*)(ef + 4 * k + h2);
#pragma unroll
      for (int n = 0; n < 4; ++n)
        acc[n] = wmma4(a, ldBp(pWskip0, 64, k, h1, n * 16 + l15), acc[n]);
    }
#pragma unroll
    for (int n = 0; n < 4; ++n) {
      float* o = out + (long)(e0 + mrow) * DIMF + n * 16 + l15;
      if (full) {
#pragma unroll
        for (int j = 0; j < 8; ++j) o[(long)j * DIMF] = acc[n][j];
      } else {
#pragma unroll
        for (int j = 0; j < 8; ++j)
          if (e0 + j + mrow < n_edges) o[(long)j * DIMF] = acc[n][j];
      }
    }
  }

  // vector out[:, 64:160] per component: V_i@Wc1 + v_feat_i@Wskip1
#pragma unroll 1
  for (int i = 0; i < 3; ++i) {
    const float* Vp = VT + i * 16 * S_LD;
    v8f acc[2] = {};
#pragma unroll
    for (int k = 0; k < 24; ++k) {
      v2f a = *(const v2f*)(Vp + l15 * S_LD + 4 * k + h2);
#pragma unroll
      for (int n = 0; n < 2; ++n)
        acc[n] = wmma4(a, ldBp(pWc1, 32, k, h1, n * 16 + l15), acc[n]);
    }
#pragma unroll
    for (int k = 0; k < 8; ++k) {
      int c = 4 * k + h2;
      v2f a;
      a.x = ef[64 + 3 * c + i];
      a.y = ef[64 + 3 * (c + 1) + i];
#pragma unroll
      for (int n = 0; n < 2; ++n)
        acc[n] = wmma4(a, ldBp(pWskip1, 32, k, h1, n * 16 + l15), acc[n]);
    }
#pragma unroll
    for (int n = 0; n < 2; ++n) {
      float* o = out + (long)(e0 + mrow) * DIMF + 64 + 3 * (n * 16 + l15) + i;
      if (full) {
#pragma unroll
        for (int j = 0; j < 8; ++j) o[(long)j * DIMF] = acc[n][j];
      } else {
#pragma unroll
        for (int j = 0; j < 8; ++j)
          if (e0 + j + mrow < n_edges) o[(long)j * DIMF] = acc[n][j];
      }
    }
  }
}

// ============================================================================
extern "C" void kernel_launch(void* const* d_in, const int* in_sizes, int n_in,
                              void* d_out, int out_size, void* d_ws, size_t ws_size,
                              hipStream_t stream) {
  const int*   edge_index = (const int*)d_in[0];
  const float* node_feats = (const float*)d_in[1];
  const float* edge_feats = (const float*)d_in[2];
  const float* edge_attrs = (const float*)d_in[3];
  const float* edge_embed = (const float*)d_in[4];
  const float* Wsrc0 = (const float*)d_in[5];
  const float* Wsrc1 = (const float*)d_in[6];
  const float* Wtar0 = (const float*)d_in[7];
  const float* Wtar1 = (const float*)d_in[8];
  const float* Wskip0 = (const float*)d_in[9];
  const float* Wskip1 = (const float*)d_in[10];
  const float* Wm1 = (const float*)d_in[11];
  const float* Wm2 = (const float*)d_in[12];
  const float* Wm3 = (const float*)d_in[13];
  const float* Wm4 = (const float*)d_in[14];
  const float* Wc0 = (const float*)d_in[15];
  const float* Wc1 = (const float*)d_in[16];

  const int n_nodes = in_sizes[1] / DIMF;
  const int n_edges = in_sizes[0] / 2;

  // workspace layout
  float* up_src = (float*)d_ws;
  float* up_tar = up_src + (size_t)n_nodes * DIMF;
  float* p = up_tar + (size_t)n_nodes * DIMF;
  float* pWsrc0  = p; p += 64 * 64;
  float* pWsrc1  = p; p += 32 * 32;
  float* pWtar0  = p; p += 64 * 64;
  float* pWtar1  = p; p += 32 * 32;
  float* pWskip0 = p; p += 64 * 64;
  float* pWskip1 = p; p += 32 * 32;
  float* pWm1    = p; p += 8 * 64;
  float* pWm2    = p; p += 64 * 64;
  float* pWm3    = p; p += 64 * 64;
  float* pWm4    = p; p += 64 * 576;
  float* pWc0    = p; p += 96 * 64;
  float* pWc1    = p; p += 96 * 32;

  const float s8    = 0.125f;                 // 1/sqrt(64)
  const float s32   = 0.17677669529663687f;   // 1/sqrt(32)
  const float s96   = 0.10206207261596575f;   // 1/sqrt(96)

  auto pack = [&](const float* src, float* dst, int K, int N, float scale) {
    int n = K * N;
    pack_b_kernel<<<(n + 255) / 256, 256, 0, stream>>>(src, dst, K, N, scale);
  };
  pack(Wsrc0, pWsrc0, 64, 64, s8);
  pack(Wsrc1, pWsrc1, 32, 32, s32);
  pack(Wtar0, pWtar0, 64, 64, s8);
  pack(Wtar1, pWtar1, 32, 32, s32);
  pack(Wskip0, pWskip0, 64, 64, s8);
  pack(Wskip1, pWskip1, 32, 32, s32);
  pack(Wm1, pWm1, 8, 64, 1.0f);
  pack(Wm2, pWm2, 64, 64, 1.0f);
  pack(Wm3, pWm3, 64, 64, 1.0f);
  pack(Wm4, pWm4, 64, 576, 1.0f);
  pack(Wc0, pWc0, 96, 64, s96);
  pack(Wc1, pWc1, 96, 32, s96);

  {
    const int wavesPerBlock = 4;
    int tiles = (n_nodes + 15) / 16;
    int blocks = (tiles + wavesPerBlock - 1) / wavesPerBlock;
    node_transform_kernel<<<blocks, wavesPerBlock * 32, 0, stream>>>(
        node_feats, pWsrc0, pWsrc1, pWtar0, pWtar1, up_src, up_tar, n_nodes);
  }
  {
    int blocks = (n_edges + 15) / 16;
    edge_fused_kernel<<<blocks, 32, 0, stream>>>(
        edge_index, edge_feats, edge_attrs, edge_embed, pWskip0, pWskip1,
        pWm1, pWm2, pWm3, pWm4, pWc0, pWc1, up_src, up_tar,
        (float*)d_out, n_edges);
  }
}